// RandomResizedCropPair_88390426952149
// MI455X (gfx1250) — compile-verified
//
#include <hip/hip_runtime.h>
#include <cstdint>

#define OUT_SZ 384
#define H_IN 512
#define W_IN 512
#define B_N 32
#define C_N 3

// One block = one output row (all 3 channels + mask) of one image.
// Stage the (up to) 7 needed input row segments into LDS with CDNA5
// async global->LDS copies, then do horizontal bilinear from LDS.
__global__ __launch_bounds__(512)
void RandomResizedCropPair_kernel(const float* __restrict__ image,
                                  const float* __restrict__ mask,
                                  const int* __restrict__ crop_size,
                                  const int* __restrict__ top,
                                  const int* __restrict__ left,
                                  float* __restrict__ out)
{
    // [0..5]: channel c rows y0/y1 ; [6]: mask row (nearest y)
    __shared__ float srow[7][W_IN];

    const int y = blockIdx.x;   // output row
    const int b = blockIdx.y;   // image index
    const int t = threadIdx.x;  // 0..511

    const int   cs  = crop_size[b];
    const int   tp  = top[b];
    const int   lf  = left[b];
    const float csf = (float)cs;
    const float s   = csf / (float)OUT_SZ;   // matches reference: (csf/SIZE)

    // ---- vertical coords (uniform across the block) ----
    float cy = ((float)y + 0.5f) * s - 0.5f;
    cy = fminf(fmaxf(cy, 0.0f), csf - 1.0f);
    const int   iy0 = (int)floorf(cy);
    const int   iy1 = min(iy0 + 1, cs - 1);
    const float wy  = cy - (float)iy0;
    const int   y0  = iy0 + tp;
    const int   y1  = iy1 + tp;
    // nearest-neighbor y for mask: floor(y * cs / SIZE)
    const int   my  = min((int)floorf((float)y * csf / (float)OUT_SZ), cs - 1) + tp;

    // ---- async stage: 7 row segments [left, left+cs) -> LDS ----
    // Clamp lanes >= cs to cs-1 (duplicate harmless; keeps EXEC all-ones).
    const int      g    = (t < cs) ? t : (cs - 1);
    const uint32_t voff = (uint32_t)((lf + g) * 4);  // per-lane global byte offset

#pragma unroll
    for (int c = 0; c < C_N; ++c) {
        const float* r0 = image + (size_t)((b * C_N + c) * H_IN + y0) * W_IN;
        const float* r1 = image + (size_t)((b * C_N + c) * H_IN + y1) * W_IN;
        uint32_t l0 = (uint32_t)(uintptr_t)(&srow[2 * c + 0][t]);
        uint32_t l1 = (uint32_t)(uintptr_t)(&srow[2 * c + 1][t]);
        asm volatile("global_load_async_to_lds_b32 %0, %1, %2"
                     :: "v"(l0), "v"(voff), "s"(r0) : "memory");
        asm volatile("global_load_async_to_lds_b32 %0, %1, %2"
                     :: "v"(l1), "v"(voff), "s"(r1) : "memory");
    }
    {
        const float* rm = mask + (size_t)(b * H_IN + my) * W_IN;
        uint32_t lm = (uint32_t)(uintptr_t)(&srow[6][t]);
        asm volatile("global_load_async_to_lds_b32 %0, %1, %2"
                     :: "v"(lm), "v"(voff), "s"(rm) : "memory");
    }

    asm volatile("s_wait_asynccnt 0" ::: "memory");
    __syncthreads();

    // ---- horizontal interpolation from LDS ----
    if (t < OUT_SZ) {
        float cx = ((float)t + 0.5f) * s - 0.5f;
        cx = fminf(fmaxf(cx, 0.0f), csf - 1.0f);
        const int   jx0 = (int)floorf(cx);        // LDS-relative (row starts at `left`)
        const int   jx1 = min(jx0 + 1, cs - 1);
        const float wx  = cx - (float)jx0;

        const float w00 = (1.0f - wy) * (1.0f - wx);
        const float w01 = (1.0f - wy) * wx;
        const float w10 = wy * (1.0f - wx);
        const float w11 = wy * wx;

#pragma unroll
        for (int c = 0; c < C_N; ++c) {
            const float v = srow[2 * c + 0][jx0] * w00 + srow[2 * c + 0][jx1] * w01
                          + srow[2 * c + 1][jx0] * w10 + srow[2 * c + 1][jx1] * w11;
            out[((size_t)(b * C_N + c) * OUT_SZ + y) * OUT_SZ + t] = v;
        }

        const int jm = min((int)floorf((float)t * csf / (float)OUT_SZ), cs - 1);
        const size_t mask_base = (size_t)B_N * C_N * OUT_SZ * OUT_SZ;
        out[mask_base + ((size_t)(b * OUT_SZ + y)) * OUT_SZ + t] = srow[6][jm];
    }
}

extern "C" void kernel_launch(void* const* d_in, const int* in_sizes, int n_in,
                              void* d_out, int out_size, void* d_ws, size_t ws_size,
                              hipStream_t stream) {
    const float* image     = (const float*)d_in[0];
    const float* mask      = (const float*)d_in[1];
    const int*   crop_size = (const int*)d_in[2];
    const int*   top       = (const int*)d_in[3];
    const int*   left      = (const int*)d_in[4];
    float*       out       = (float*)d_out;

    dim3 grid(OUT_SZ, B_N);  // 384 rows x 32 images
    dim3 block(512);
    RandomResizedCropPair_kernel<<<grid, block, 0, stream>>>(
        image, mask, crop_size, top, left, out);
}